// AggregationLayer2_90950227460655
// MI455X (gfx1250) — compile-verified
//
#include <hip/hip_runtime.h>

// CDNA5 / gfx1250: fp32 WMMA path (V_WMMA_F32_16X16X4_F32), wave32.
// Local 5x5 attention aggregation as a banded 16x20 x 20x64 matmul per
// (tile-column, dj), staged through LDS so every global byte moves once.
// cur/self-weight prefetched ahead of the WMMA loops to hide HBM latency.

typedef __attribute__((ext_vector_type(2))) float v2f;
typedef __attribute__((ext_vector_type(4))) float v4f;
typedef __attribute__((ext_vector_type(8))) float v8f;

#define KS    5                 // 5x5 window
#define HALO  2
#define TILE  16                // 16x16 pixel tile per workgroup
#define RT    (TILE + 2*HALO)   // 20 (tile + halo)
#define BINS  64
#define NSLOT 26                // 25 neighbors + self
#define H_    128
#define W_    128
#define B_    4

__global__ __launch_bounds__(256) void agg_wmma_kernel(
    const float* __restrict__ attn,   // [B,H,W,26]
    const float* __restrict__ refv,   // [B,H,W,64]
    const float* __restrict__ cur,    // [B,H,W,64]
    float* __restrict__ out)          // [B,H,W,64]
{
  extern __shared__ float lds[];
  float* ref_lds  = lds;                        // [RT][RT][BINS]  = 25600 f32
  float* attn_lds = lds + RT * RT * BINS;       // [TILE][TILE][NSLOT] = 6656 f32

  const int tid = threadIdx.x;
  const int wg  = blockIdx.x;
  const int tiles_w = W_ / TILE;                // 8
  const int tiles_h = H_ / TILE;                // 8
  const int b  = wg / (tiles_w * tiles_h);
  const int tr = (wg / tiles_w) % tiles_h;
  const int tc = wg % tiles_w;
  const int i0 = tr * TILE;
  const int j0 = tc * TILE;

  // ---- Stage ref halo tile into LDS (zero-padded OOB). 6400 f32x4 / 256 thr = 25 each.
  for (int it = 0; it < 25; ++it) {
    const int e    = tid + it * 256;
    const int bin4 = e & 15;        // 16 float4 per pixel (64 bins)
    const int rem  = e >> 4;
    const int jj   = rem % RT;
    const int r    = rem / RT;
    const int gi   = i0 - HALO + r;
    const int gj   = j0 - HALO + jj;
    v4f val = {0.f, 0.f, 0.f, 0.f};
    if (gi >= 0 && gi < H_ && gj >= 0 && gj < W_) {
      const v4f* p = (const v4f*)(refv + (((size_t)b * H_ + gi) * W_ + gj) * BINS);
      val = p[bin4];
    }
    *(v4f*)(ref_lds + ((size_t)r * RT + jj) * BINS + bin4 * 4) = val;
  }

  // ---- Stage attn tile into LDS. 6656 f32 = 26 * 256 exactly.
  for (int it = 0; it < 26; ++it) {
    const int e   = tid + it * 256;
    const int k   = e % NSLOT;
    const int rem = e / NSLOT;
    const int j   = rem & 15;
    const int m   = rem >> 4;
    attn_lds[e] = attn[(((size_t)b * H_ + (i0 + m)) * W_ + (j0 + j)) * NSLOT + k];
  }
  __syncthreads();

  const int wave  = tid >> 5;
  const int lane  = tid & 31;
  const int mrow  = lane & 15;   // M (A) / N (B,C) index
  const int khalf = lane >> 4;   // which K pair this half-wave owns

  // 16 tile columns / 8 waves = 2 columns per wave. Each column computes all
  // 4 bin-blocks at once: A fragments built once, reused by 4 WMMAs into 4
  // independent accumulators (no back-to-back accumulation dependency).
  for (int colIdx = 0; colIdx < 2; ++colIdx) {
    const int j  = wave * 2 + colIdx;  // tile column 0..15
    const int gj = j0 + j;

    // ---- Prefetch epilogue operands BEFORE the matrix loops so the global
    // loads of `cur` overlap the 100-WMMA chain instead of stalling at the end.
    float curv[32];
    float aself[8];
    #pragma unroll
    for (int v = 0; v < 8; ++v) {
      const int p = v + 8 * khalf;
      const size_t base = (((size_t)b * H_ + (i0 + p)) * W_ + gj) * BINS + mrow;
      aself[v] = attn_lds[((size_t)p * TILE + j) * NSLOT + (NSLOT - 1)];
      curv[v * 4 + 0] = cur[base +  0];
      curv[v * 4 + 1] = cur[base + 16];
      curv[v * 4 + 2] = cur[base + 32];
      curv[v * 4 + 3] = cur[base + 48];
    }

    v8f acc0 = {}, acc1 = {}, acc2 = {}, acc3 = {};

    const float* arow = attn_lds + ((size_t)mrow * TILE + j) * NSLOT;

    #pragma unroll
    for (int dj = 0; dj < KS; ++dj) {
      const int jj = j + dj;           // padded LDS column index (0..19)
      #pragma unroll
      for (int kk = 0; kk < 5; ++kk) { // K = 20 halo rows, 5 steps of 4
        const int rA0 = kk * 4 + 2 * khalf;  // this half-wave's K indices
        const int d0 = rA0 - mrow, d1 = d0 + 1;

        // Banded A fragment (16x4 fp32): A[m,r] = attn[m,(r-m)*5+dj] iff 0<=r-m<5
        v2f a;
        a.x = (d0 >= 0 && d0 < KS) ? arow[d0 * KS + dj] : 0.f;
        a.y = (d1 >= 0 && d1 < KS) ? arow[d1 * KS + dj] : 0.f;

        // 4 B fragments (4x16 fp32 each), one per 16-bin block.
        const float* bbase = ref_lds + ((size_t)rA0 * RT + jj) * BINS + mrow;
        v2f b0, b1, b2, b3;
        b0.x = bbase[0];  b0.y = bbase[(size_t)RT * BINS];
        b1.x = bbase[16]; b1.y = bbase[(size_t)RT * BINS + 16];
        b2.x = bbase[32]; b2.y = bbase[(size_t)RT * BINS + 32];
        b3.x = bbase[48]; b3.y = bbase[(size_t)RT * BINS + 48];

        acc0 = __builtin_amdgcn_wmma_f32_16x16x4_f32(false, a, false, b0, (short)0, acc0, false, false);
        acc1 = __builtin_amdgcn_wmma_f32_16x16x4_f32(false, a, false, b1, (short)0, acc1, false, false);
        acc2 = __builtin_amdgcn_wmma_f32_16x16x4_f32(false, a, false, b2, (short)0, acc2, false, false);
        acc3 = __builtin_amdgcn_wmma_f32_16x16x4_f32(false, a, false, b3, (short)0, acc3, false, false);
      }
    }

    // ---- Epilogue: pure FMA + store (cur already in registers).
    // C layout: VGPR v, lanes 0-15 -> (M=v, N=lane); lanes 16-31 -> (M=v+8, N=lane-16)
    #pragma unroll
    for (int v = 0; v < 8; ++v) {
      const int p = v + 8 * khalf;
      const size_t base = (((size_t)b * H_ + (i0 + p)) * W_ + gj) * BINS + mrow;
      out[base +  0] = acc0[v] + aself[v] * curv[v * 4 + 0];
      out[base + 16] = acc1[v] + aself[v] * curv[v * 4 + 1];
      out[base + 32] = acc2[v] + aself[v] * curv[v * 4 + 2];
      out[base + 48] = acc3[v] + aself[v] * curv[v * 4 + 3];
    }
  }
}

extern "C" void kernel_launch(void* const* d_in, const int* in_sizes, int n_in,
                              void* d_out, int out_size, void* d_ws, size_t ws_size,
                              hipStream_t stream) {
  (void)in_sizes; (void)n_in; (void)out_size; (void)d_ws; (void)ws_size;
  const float* attn = (const float*)d_in[0];
  const float* refv = (const float*)d_in[1];
  const float* cur  = (const float*)d_in[2];
  float* out = (float*)d_out;

  const size_t shmem = (size_t)(RT * RT * BINS + TILE * TILE * NSLOT) * sizeof(float);
  const int nblocks = B_ * (H_ / TILE) * (W_ / TILE);  // 256 workgroups
  agg_wmma_kernel<<<dim3(nblocks), dim3(256), shmem, stream>>>(attn, refv, cur, out);
}